// RNNAttention_55001351192736
// MI455X (gfx1250) — compile-verified
//
#include <hip/hip_runtime.h>
#include <hip/hip_bf16.h>
#include <math.h>

// RNN + Bahdanau attention + vocab projection for MI455X (gfx1250).
// Dominant cost: [1024x256] @ [256x32000] fp32 GEMM -> V_WMMA_F32_16X16X4_F32.

typedef __attribute__((ext_vector_type(2))) float v2f;
typedef __attribute__((ext_vector_type(8))) float v8f;

#define LSEQ 1024
#define HDIM 128
#define VOCAB 32000

// ---------------------------------------------------------------------------
// Generic fp32 WMMA GEMM:  C[M x N] = A[M x K] * B + bias
//   BT=false: B is row-major [K x N] (ldb)
//   BT=true : B is row-major [N x K] (ldb)  (i.e. C = A @ B^T)
//   Arows   : optional row-gather indices for A (embedding lookup)
//   Mstore  : only rows < Mstore are written (drops last logits row)
// Macro tile 64x64, KC=64, 256 threads = 8 waves; each wave owns two 16x16
// accumulators (n and n+32). LDS pitch 65 => conflict-free fragment reads.
// ---------------------------------------------------------------------------
template <bool BT>
__global__ __launch_bounds__(256)
void wmma_gemm_f32(const float* __restrict__ A, const int* __restrict__ Arows, int lda,
                   const float* __restrict__ B, int ldb,
                   const float* __restrict__ bias,
                   float* __restrict__ C, int ldc,
                   int K, int Mstore)
{
    __shared__ float As[64][65];  // [m][k]
    __shared__ float Bs[64][65];  // [k][n]

    const int tid  = threadIdx.x;
    const int lane = tid & 31;
    const int wave = tid >> 5;     // 0..7
    const int wm   = wave >> 1;    // 0..3 : 16-row sub-tile
    const int wn   = wave & 1;     // 0..1 : 16-col sub-tile (covers n, n+32)
    const int m0   = blockIdx.y * 64;
    const int n0   = blockIdx.x * 64;
    const int half = lane >> 4;    // 0: K{0,1}/M+0 ; 1: K{2,3}/M+8
    const int l16  = lane & 15;

    v8f c0 = {};
    v8f c1 = {};

    for (int k0 = 0; k0 < K; k0 += 64) {
        // ---- stage A chunk: 64 rows x 64 k ----
        {
            const int r    = tid >> 2;                  // 0..63
            const int cthr = tid & 3;
            const int grow = Arows ? Arows[m0 + r] : (m0 + r);
            const float* src = A + (size_t)grow * lda + k0;
#pragma unroll
            for (int u = 0; u < 4; ++u) {
                const int kk = (cthr + u * 4) * 4;
                const float4 v = *(const float4*)(src + kk);
                As[r][kk + 0] = v.x; As[r][kk + 1] = v.y;
                As[r][kk + 2] = v.z; As[r][kk + 3] = v.w;
            }
        }
        // ---- stage B chunk into [k][n] layout ----
        if (!BT) {
            const int kk   = tid >> 2;
            const int cthr = tid & 3;
            const float* src = B + (size_t)(k0 + kk) * ldb + n0;
#pragma unroll
            for (int u = 0; u < 4; ++u) {
                const int n = (cthr + u * 4) * 4;
                const float4 v = *(const float4*)(src + n);
                Bs[kk][n + 0] = v.x; Bs[kk][n + 1] = v.y;
                Bs[kk][n + 2] = v.z; Bs[kk][n + 3] = v.w;
            }
        } else {
            const int n    = tid >> 2;
            const int cthr = tid & 3;
            const float* src = B + (size_t)(n0 + n) * ldb + k0;
#pragma unroll
            for (int u = 0; u < 4; ++u) {
                const int kk = (cthr + u * 4) * 4;
                const float4 v = *(const float4*)(src + kk);
                Bs[kk + 0][n] = v.x; Bs[kk + 1][n] = v.y;
                Bs[kk + 2][n] = v.z; Bs[kk + 3][n] = v.w;
            }
        }
        __syncthreads();

        // ---- 16 x (16x16x4) WMMA steps over the K chunk ----
#pragma unroll
        for (int k4 = 0; k4 < 64; k4 += 4) {
            const int ka = k4 + half * 2;  // lanes 0-15: K0,K1 ; lanes 16-31: K2,K3
            v2f a, b0, b1;
            a[0]  = As[wm * 16 + l16][ka];
            a[1]  = As[wm * 16 + l16][ka + 1];
            b0[0] = Bs[ka][wn * 16 + l16];
            b0[1] = Bs[ka + 1][wn * 16 + l16];
            b1[0] = Bs[ka][wn * 16 + 32 + l16];
            b1[1] = Bs[ka + 1][wn * 16 + 32 + l16];
            c0 = __builtin_amdgcn_wmma_f32_16x16x4_f32(false, a, false, b0,
                                                       (short)0, c0, false, false);
            c1 = __builtin_amdgcn_wmma_f32_16x16x4_f32(false, a, false, b1,
                                                       (short)0, c1, false, false);
        }
        __syncthreads();
    }

    // ---- epilogue: C/D layout -> lane l16 = column, vgpr i = M (half*8 + i) ----
    const int ncol0 = n0 + wn * 16 + l16;
    const int ncol1 = ncol0 + 32;
    const float add0 = bias ? bias[ncol0] : 0.0f;
    const float add1 = bias ? bias[ncol1] : 0.0f;
#pragma unroll
    for (int i = 0; i < 8; ++i) {
        const int row = m0 + wm * 16 + half * 8 + i;
        if (row < Mstore) {
            C[(size_t)row * ldc + ncol0] = c0[i] + add0;
            C[(size_t)row * ldc + ncol1] = c1[i] + add1;
        }
    }
}

// ---------------------------------------------------------------------------
// Sequential RNN scan: h_t = tanh(inp_t + w_hh @ h_{t-1} + b_hh)
// One workgroup, 256 threads: thread (i, half) keeps w_hh[i][half*64..+63]
// in 64 VGPRs; h broadcast from LDS; split-K partials combined through LDS.
// ---------------------------------------------------------------------------
__global__ __launch_bounds__(256)
void rnn_scan_kernel(const float* __restrict__ inp, const float* __restrict__ w_hh,
                     const float* __restrict__ b_hh, float* __restrict__ out)
{
    __shared__ float hbuf[HDIM];
    __shared__ float partial[HDIM];
    const int tid  = threadIdx.x;
    const int i    = tid & (HDIM - 1);
    const int half = tid >> 7;

    float w[64];
#pragma unroll
    for (int j = 0; j < 64; ++j)
        w[j] = w_hh[i * HDIM + half * 64 + j];
    const float bias = b_hh[i];

    if (tid < HDIM) hbuf[tid] = 0.0f;
    __syncthreads();

    for (int t = 0; t < LSEQ; ++t) {
        float acc = 0.0f;
#pragma unroll
        for (int j = 0; j < 64; j += 4) {
            const float4 h4 = *(const float4*)&hbuf[half * 64 + j];  // LDS broadcast
            acc = fmaf(w[j + 0], h4.x, acc);
            acc = fmaf(w[j + 1], h4.y, acc);
            acc = fmaf(w[j + 2], h4.z, acc);
            acc = fmaf(w[j + 3], h4.w, acc);
        }
        if (half) partial[i] = acc;
        __syncthreads();
        if (!half) {
            const float hn = tanhf(acc + partial[i] + inp[t * HDIM + i] + bias);
            out[t * HDIM + i] = hn;
            hbuf[i] = hn;
        }
        __syncthreads();
    }
}

// ---------------------------------------------------------------------------
// scores[t][s] = (1/sqrt(H)) * sum_h v[h] * tanh(q[t][h] + k[s][h]), s <= t
// 32x32 (t,s) tiles; fully-masked tiles exit early.
// ---------------------------------------------------------------------------
__global__ __launch_bounds__(256)
void attn_scores_kernel(const float* __restrict__ qrys, const float* __restrict__ keys,
                        const float* __restrict__ vvec, float* __restrict__ scores)
{
    const int t0 = blockIdx.y * 32;
    const int s0 = blockIdx.x * 32;
    if (s0 > t0 + 31) return;  // entire tile above the causal diagonal

    __shared__ float qs[32][129];
    __shared__ float ks[32][129];
    __shared__ float vs[HDIM];
    const int tid = threadIdx.x;

    {
        const int r = tid >> 3;          // 32 rows, 8 threads/row
        const int c = (tid & 7) * 16;    // 16 floats = 4 float4 per thread
#pragma unroll
        for (int u = 0; u < 4; ++u) {
            const float4 a = *(const float4*)(qrys + (t0 + r) * HDIM + c + u * 4);
            qs[r][c + u * 4 + 0] = a.x; qs[r][c + u * 4 + 1] = a.y;
            qs[r][c + u * 4 + 2] = a.z; qs[r][c + u * 4 + 3] = a.w;
            const float4 b = *(const float4*)(keys + (s0 + r) * HDIM + c + u * 4);
            ks[r][c + u * 4 + 0] = b.x; ks[r][c + u * 4 + 1] = b.y;
            ks[r][c + u * 4 + 2] = b.z; ks[r][c + u * 4 + 3] = b.w;
        }
    }
    if (tid < 32) {
        const float4 v = *(const float4*)(vvec + tid * 4);
        vs[tid * 4 + 0] = v.x; vs[tid * 4 + 1] = v.y;
        vs[tid * 4 + 2] = v.z; vs[tid * 4 + 3] = v.w;
    }
    __syncthreads();

    const float inv_sqrt_h = 0.08838834764831844f;  // 1/sqrt(128)
#pragma unroll
    for (int pp = 0; pp < 4; ++pp) {
        const int p  = tid + 256 * pp;
        const int tl = p >> 5;
        const int sl = p & 31;
        const int t  = t0 + tl;
        const int s  = s0 + sl;
        if (s > t) continue;
        float e = 0.0f;
        for (int h = 0; h < HDIM; ++h)
            e += vs[h] * tanhf(qs[tl][h] + ks[sl][h]);
        scores[t * LSEQ + s] = e * inv_sqrt_h;
    }
}

// ---------------------------------------------------------------------------
// Causal softmax per row; writes 0 above the diagonal so ctx GEMM is dense.
// ---------------------------------------------------------------------------
__global__ __launch_bounds__(256)
void softmax_causal_kernel(float* __restrict__ scores)
{
    const int t = blockIdx.x;
    const int n = t + 1;
    float* row = scores + (size_t)t * LSEQ;
    const int tid = threadIdx.x;
    __shared__ float red[256];

    float m = -__builtin_inff();
    for (int s = tid; s < n; s += 256) m = fmaxf(m, row[s]);
    red[tid] = m; __syncthreads();
    for (int off = 128; off; off >>= 1) {
        if (tid < off) red[tid] = fmaxf(red[tid], red[tid + off]);
        __syncthreads();
    }
    m = red[0]; __syncthreads();

    float sum = 0.0f;
    for (int s = tid; s < n; s += 256) sum += expf(row[s] - m);
    red[tid] = sum; __syncthreads();
    for (int off = 128; off; off >>= 1) {
        if (tid < off) red[tid] += red[tid + off];
        __syncthreads();
    }
    const float invs = 1.0f / red[0];

    for (int s = tid; s < LSEQ; s += 256)
        row[s] = (s < n) ? expf(row[s] - m) * invs : 0.0f;
}

// ---------------------------------------------------------------------------
// combined[t] = concat(out[t], ctx[t])
// ---------------------------------------------------------------------------
__global__ __launch_bounds__(256)
void pack_combined_kernel(const float* __restrict__ out, const float* __restrict__ ctx,
                          float* __restrict__ comb)
{
    const int idx = blockIdx.x * 256 + threadIdx.x;  // over 1024*128
    const int t = idx >> 7;
    const int i = idx & 127;
    comb[t * 256 + i]       = out[idx];
    comb[t * 256 + 128 + i] = ctx[idx];
}

// ---------------------------------------------------------------------------
extern "C" void kernel_launch(void* const* d_in, const int* in_sizes, int n_in,
                              void* d_out, int out_size, void* d_ws, size_t ws_size,
                              hipStream_t stream)
{
    (void)in_sizes; (void)n_in; (void)out_size; (void)ws_size;

    const int*   x        = (const int*)  d_in[0];   // (1, L)
    const float* emb_tab  = (const float*)d_in[1];   // (V, E)
    const float* w_ih     = (const float*)d_in[2];   // (H, E)
    const float* w_hh     = (const float*)d_in[3];   // (H, H)
    const float* b_ih     = (const float*)d_in[4];   // (H,)
    const float* b_hh     = (const float*)d_in[5];   // (H,)
    const float* attn_w_w = (const float*)d_in[6];   // (H, 2H)  [Wk | Wq]
    const float* attn_w_b = (const float*)d_in[7];   // (H,)
    const float* attn_v_w = (const float*)d_in[8];   // (1, H)
    const float* fc_w     = (const float*)d_in[9];   // (V, 2H)
    const float* fc_b     = (const float*)d_in[10];  // (V,)
    float* logits = (float*)d_out;                   // (L-1, V)

    float* ws   = (float*)d_ws;
    float* inp  = ws;                        // 1024*128
    float* hout = inp  + LSEQ * HDIM;        // 1024*128
    float* keys = hout + LSEQ * HDIM;        // 1024*128
    float* qrys = keys + LSEQ * HDIM;        // 1024*128
    float* scor = qrys + LSEQ * HDIM;        // 1024*1024
    float* ctx  = scor + LSEQ * LSEQ;        // 1024*128
    float* comb = ctx  + LSEQ * HDIM;        // 1024*256

    const dim3 blk(256);

    // 1. inp = emb_table[x] @ w_ih^T + b_ih          (gather-A, B^T)
    wmma_gemm_f32<true><<<dim3(HDIM / 64, LSEQ / 64), blk, 0, stream>>>(
        emb_tab, x, 128, w_ih, 128, b_ih, inp, HDIM, 128, LSEQ);

    // 2. sequential RNN scan
    rnn_scan_kernel<<<1, 256, 0, stream>>>(inp, w_hh, b_hh, hout);

    // 3. keys = out @ Wk^T   (Wk = attn_w_w[:, :H], row pitch 2H)
    wmma_gemm_f32<true><<<dim3(HDIM / 64, LSEQ / 64), blk, 0, stream>>>(
        hout, nullptr, HDIM, attn_w_w, 2 * HDIM, nullptr, keys, HDIM, HDIM, LSEQ);

    // 4. qrys = out @ Wq^T + b   (Wq = attn_w_w[:, H:])
    wmma_gemm_f32<true><<<dim3(HDIM / 64, LSEQ / 64), blk, 0, stream>>>(
        hout, nullptr, HDIM, attn_w_w + HDIM, 2 * HDIM, attn_w_b, qrys, HDIM, HDIM, LSEQ);

    // 5. additive-attention scores (causal tiles only)
    attn_scores_kernel<<<dim3(LSEQ / 32, LSEQ / 32), blk, 0, stream>>>(
        qrys, keys, attn_v_w, scor);

    // 6. causal softmax (zero-fills above diagonal)
    softmax_causal_kernel<<<LSEQ, blk, 0, stream>>>(scor);

    // 7. ctx = attn @ out   (dense GEMM, B row-major)
    wmma_gemm_f32<false><<<dim3(HDIM / 64, LSEQ / 64), blk, 0, stream>>>(
        scor, nullptr, LSEQ, hout, HDIM, nullptr, ctx, HDIM, LSEQ, LSEQ);

    // 8. combined = [out | ctx]
    pack_combined_kernel<<<(LSEQ * HDIM) / 256, blk, 0, stream>>>(hout, ctx, comb);

    // 9. logits = combined @ fc_w^T + fc_b   (store only first L-1 rows)
    wmma_gemm_f32<true><<<dim3(VOCAB / 64, LSEQ / 64), blk, 0, stream>>>(
        comb, nullptr, 2 * HDIM, fc_w, 2 * HDIM, fc_b, logits, VOCAB, 2 * HDIM, LSEQ - 1);
}